// SpatialTransformerGauss3d_9792525435348
// MI455X (gfx1250) — compile-verified
//
#include <hip/hip_runtime.h>
#include <hip/hip_bf16.h>

// ---------------------------------------------------------------------------
// SpatialTransformerGauss3d for MI455X (gfx1250, wave32).
//
// Memory-bound op: x = 64 MB, pyramid workspace ~89 MB (fits 192 MB L2),
// 21M bilinear-corner gathers. Pyramid stored CHANNEL-LAST so one corner
// gather = one aligned 64B read covering all 16 channels.
// Gaussian blur done as fp32 banded GEMM lo = B*X*B (B symmetric tridiag
// 1/4,1/2,1/4) using V_WMMA_F32_16X16X4_F32 -> exact fp32 numerics.
// Image load into LDS uses GLOBAL_LOAD_ASYNC_TO_LDS_B128 (ASYNCcnt path),
// overlapping with band-matrix construction in LDS.
// ---------------------------------------------------------------------------

typedef __attribute__((ext_vector_type(2))) float v2f;
typedef __attribute__((ext_vector_type(8))) float v8f;

// Pointer types for the async global->LDS builtin (per hipcc diagnostic:
// param is 'int __vector_size__(16) __device__ *').
typedef int v4i_g __attribute__((vector_size(16)));
typedef __attribute__((address_space(1))) v4i_g* as1_v4i_ptr;
typedef __attribute__((address_space(3))) v4i_g* as3_v4i_ptr;

#define SX 68                 // LDS row stride (floats): 68*4B keeps 16B row
                              // alignment and offsets banks by 4 per row.
#define WS_PER_BATCH 87296    // 64^2*16 + 32^2*16 + 16^2*16 + 8^2*16 + 4^2*16

#if defined(__has_builtin)
#if __has_builtin(__builtin_amdgcn_global_load_async_to_lds_b128) && \
    __has_builtin(__builtin_amdgcn_s_wait_asynccnt)
#define HAVE_ASYNC_LDS 1
#endif
#endif
#ifndef HAVE_ASYNC_LDS
#define HAVE_ASYNC_LDS 0
#endif

// One 16x16 fp32 output tile of (Amat * Bmat), K = padN, via 16x16x4 WMMA.
// Amat rows at stride SX, Bmat rows at stride SX.
// rowA = tile-row*16 + lane%16 ; colB = tile-col*16 + lane%16 ;
// kOff = 2*(lane/16).
__device__ inline v8f wmma_tile16(const float* __restrict__ Amat,
                                  const float* __restrict__ Bmat,
                                  int rowA, int colB, int kOff, int padN) {
  v8f acc = {0.f, 0.f, 0.f, 0.f, 0.f, 0.f, 0.f, 0.f};
  const float* ap = Amat + rowA * SX;
  for (int kc = 0; kc < padN; kc += 4) {
    const int k = kc + kOff;
    v2f a  = { ap[k], ap[k + 1] };                              // A[m][k],A[m][k+1]
    v2f bb = { Bmat[k * SX + colB], Bmat[(k + 1) * SX + colB] };// B[k][n],B[k+1][n]
    acc = __builtin_amdgcn_wmma_f32_16x16x4_f32(
        false, a, false, bb, (short)0, acc, false, false);
  }
  return acc;
}

// Kernel A: build Laplacian pyramid for one (batch, channel).
// grid = (16 channels, 256 batches), 256 threads (8 waves).
__global__ __launch_bounds__(256) void pyramid_kernel(
    const float* __restrict__ x, float* __restrict__ ws) {
  __shared__ float Xs[64 * SX];   // current level image (zero-padded to >=16)
  __shared__ float Vs[64 * SX];   // vertical blur V = B*X
  __shared__ float Bs[64 * SX];   // banded blur matrix (symmetric)
  __shared__ float Xn[32 * 34];   // downsampled next level

  const int tid  = threadIdx.x;
  const int ch   = blockIdx.x;    // 0..15
  const int b    = blockIdx.y;    // 0..255  (b = n*32 + t)
  const int nIdx = b >> 5;
  const int tt   = b & 31;

  const int lane = tid & 31;
  const int wave = tid >> 5;
  const int m16  = lane & 15;
  const int kOff = (lane >> 4) << 1;   // 2*(lane/16)
  const int rOff = (lane >> 4) << 3;   // 8*(lane/16)

  // ---- load the 64x64 source image (contiguous 16 KB) into LDS ----
  const float* src = x + ((((size_t)nIdx * 16 + ch) * 32 + tt) << 12);
#if HAVE_ASYNC_LDS
  // Async-copy engine: per-lane 16B global->LDS, tracked by ASYNCcnt.
  // Overlaps with band-matrix construction below (first loop iteration).
#pragma unroll
  for (int i = 0; i < 4; ++i) {
    int e4  = tid + i * 256;
    int row = e4 >> 4;
    int c0  = (e4 & 15) << 2;
    __builtin_amdgcn_global_load_async_to_lds_b128(
        (as1_v4i_ptr)(src + (e4 << 2)),
        (as3_v4i_ptr)(&Xs[row * SX + c0]),
        0, 0);
  }
#else
  for (int e4 = tid; e4 < 1024; e4 += 256) {
    float4 v = reinterpret_cast<const float4*>(src)[e4];
    int row = e4 >> 4;
    int c0  = (e4 & 15) << 2;
    float* d = &Xs[row * SX + c0];
    d[0] = v.x; d[1] = v.y; d[2] = v.z; d[3] = v.w;
  }
#endif

  const int   lvlOff[4] = {0, 65536, 81920, 86016};
  float* wsb = ws + (size_t)b * WS_PER_BATCH + ch;   // channel-last base

  for (int m = 0; m < 4; ++m) {
    const int n    = 64 >> m;
    const int padN = (n < 16) ? 16 : n;
    const int nt   = padN >> 4;           // tiles per dim
    const int ntt  = nt * nt;

    // build banded matrix B_n (zero outside band / outside n)
    for (int e = tid; e < padN * padN; e += 256) {
      int i = e / padN, j = e - i * padN;
      float v = 0.f;
      if (i < n && j < n) {
        int d = i - j;
        if (d == 0)                 v = 0.5f;
        else if (d == 1 || d == -1) v = 0.25f;
      }
      Bs[i * SX + j] = v;
    }
#if HAVE_ASYNC_LDS
    if (m == 0) {
      // Barrier does NOT flush ASYNCcnt: wait for this wave's async loads
      // before the workgroup barrier makes Xs globally visible.
      __builtin_amdgcn_s_wait_asynccnt(0);
    }
#endif
    __syncthreads();

    // GEMM 1: V = B * X   (vertical blur)
    for (int tile = wave; tile < ntt; tile += 8) {
      int rt = tile / nt, ct = tile - rt * nt;
      int rowA = rt * 16 + m16;
      int colB = ct * 16 + m16;
      v8f acc = wmma_tile16(Bs, Xs, rowA, colB, kOff, padN);
#pragma unroll
      for (int j = 0; j < 8; ++j)
        Vs[(rt * 16 + j + rOff) * SX + colB] = acc[j];
    }
    __syncthreads();

    // GEMM 2: LO = V * B (horizontal blur; B symmetric) + epilogue:
    // highpass = X - LO -> workspace (channel-last); downsample LO -> Xn.
    float* outL = wsb + lvlOff[m];
    for (int tile = wave; tile < ntt; tile += 8) {
      int rt = tile / nt, ct = tile - rt * nt;
      int rowA = rt * 16 + m16;
      int colB = ct * 16 + m16;
      v8f acc = wmma_tile16(Vs, Bs, rowA, colB, kOff, padN);
#pragma unroll
      for (int j = 0; j < 8; ++j) {
        int r = rt * 16 + j + rOff;
        int c = colB;
        if (r < n && c < n) {
          float lo = acc[j];
          outL[(size_t)(r * n + c) << 4] = Xs[r * SX + c] - lo;
          if (((r | c) & 1) == 0) Xn[(r >> 1) * 34 + (c >> 1)] = lo;
        }
      }
    }
    __syncthreads();

    if (m < 3) {
      const int nn = n >> 1;
      const int pp = (nn < 16) ? 16 : nn;   // zero-pad up to 16 for WMMA
      for (int e = tid; e < pp * pp; e += 256) {
        int i = e / pp, j = e - i * pp;
        Xs[i * SX + j] = (i < nn && j < nn) ? Xn[i * 34 + j] : 0.f;
      }
      __syncthreads();
    }
  }

  // final low image (4x4) -> level 4
  if (tid < 16) {
    int r = tid >> 2, c = tid & 3;
    ws[(size_t)b * WS_PER_BATCH + 87040 + ((size_t)tid << 4) + ch] =
        Xn[r * 34 + c];
  }
}

// Kernel B: bilinear sample 5 levels + weighted channel reduction.
// grid = (16 o-tiles, 256 batches), 256 threads; one thread per (b,o).
__global__ __launch_bounds__(256) void sample_kernel(
    const float* __restrict__ grid, const float* __restrict__ feats,
    const float* __restrict__ bias, const float* __restrict__ ws,
    float* __restrict__ out) {
  const int o = blockIdx.x * 256 + threadIdx.x;   // 0..4095
  const int b = blockIdx.y;                       // 0..255

  float gx = grid[2 * o];
  float gy = grid[2 * o + 1];
  gx = fminf(fmaxf(gx, -1.f), 1.f);
  gy = fminf(fmaxf(gy, -1.f), 1.f);

  const float* wb = ws + (size_t)b * WS_PER_BATCH;
  const int nlv[5] = {64, 32, 16, 8, 4};
  const int off[5] = {0, 65536, 81920, 86016, 87040};

  float acc = 0.f;
#pragma unroll
  for (int m = 0; m < 5; ++m) {
    const int n = nlv[m];
    float fx = (gx + 1.f) * (n * 0.5f) - 0.5f;
    float fy = (gy + 1.f) * (n * 0.5f) - 0.5f;
    float x0f = floorf(fx), y0f = floorf(fy);
    int   ix0 = (int)x0f,  iy0 = (int)y0f;
    float wx1 = fx - x0f, wx0 = 1.f - wx1;
    float wy1 = fy - y0f, wy0 = 1.f - wy1;

    const float* lp = wb + off[m];
    const float4* p[4];
    float w[4];
    const int   xs[2]  = {ix0, ix0 + 1};
    const int   ys[2]  = {iy0, iy0 + 1};
    const float wxs[2] = {wx0, wx1};
    const float wys[2] = {wy0, wy1};
#pragma unroll
    for (int cy = 0; cy < 2; ++cy) {
#pragma unroll
      for (int cx = 0; cx < 2; ++cx) {
        int xi = xs[cx], yi = ys[cy];
        bool valid = (xi >= 0) && (xi < n) && (yi >= 0) && (yi < n);
        int xc = xi < 0 ? 0 : (xi > n - 1 ? n - 1 : xi);
        int yc = yi < 0 ? 0 : (yi > n - 1 ? n - 1 : yi);
        int idx = cy * 2 + cx;
        p[idx] = reinterpret_cast<const float4*>(
            lp + ((size_t)(yc * n + xc) << 4));   // one 64B line = 16 ch
        w[idx] = valid ? wxs[cx] * wys[cy] : 0.f;
      }
    }

    const float* fb = feats + (size_t)(m * 16) * 4096 + o;
#pragma unroll
    for (int cc = 0; cc < 4; ++cc) {
      float4 v0 = p[0][cc], v1 = p[1][cc], v2 = p[2][cc], v3 = p[3][cc];
      float s0 = w[0] * v0.x + w[1] * v1.x + w[2] * v2.x + w[3] * v3.x;
      float s1 = w[0] * v0.y + w[1] * v1.y + w[2] * v2.y + w[3] * v3.y;
      float s2 = w[0] * v0.z + w[1] * v1.z + w[2] * v2.z + w[3] * v3.z;
      float s3 = w[0] * v0.w + w[1] * v1.w + w[2] * v2.w + w[3] * v3.w;
      float f0 = fmaxf(fb[(size_t)(cc * 4 + 0) * 4096], 0.f);
      float f1 = fmaxf(fb[(size_t)(cc * 4 + 1) * 4096], 0.f);
      float f2 = fmaxf(fb[(size_t)(cc * 4 + 2) * 4096], 0.f);
      float f3 = fmaxf(fb[(size_t)(cc * 4 + 3) * 4096], 0.f);
      acc += s0 * f0 + s1 * f1 + s2 * f2 + s3 * f3;
    }
  }

  out[(size_t)b * 4096 + o] = acc + bias[o];
}

extern "C" void kernel_launch(void* const* d_in, const int* in_sizes, int n_in,
                              void* d_out, int out_size, void* d_ws, size_t ws_size,
                              hipStream_t stream) {
  const float* x     = (const float*)d_in[0];  // (8,16,32,64,64)
  const float* grid  = (const float*)d_in[1];  // (1,4096,1,2)
  const float* feats = (const float*)d_in[2];  // (1,80,1,4096)
  const float* bias  = (const float*)d_in[3];  // (4096)
  float* out = (float*)d_out;                  // (8,32,4096)
  float* ws  = (float*)d_ws;                   // needs 256*87296*4 ~= 85.3 MiB

  // Kernel A: 16 channels x 256 batches, 8 waves each; WMMA fp32 blur GEMMs,
  // async global->LDS image staging when available.
  pyramid_kernel<<<dim3(16, 256), 256, 0, stream>>>(x, ws);
  // Kernel B: 4096 outputs x 256 batches; 64B-per-corner gathers from L2.
  sample_kernel<<<dim3(16, 256), 256, 0, stream>>>(grid, feats, bias, ws, out);
}